// Network_58282706207074
// MI455X (gfx1250) — compile-verified
//
#include <hip/hip_runtime.h>
#include <cmath>

typedef float v2f __attribute__((ext_vector_type(2)));
typedef float v8f __attribute__((ext_vector_type(8)));

#define NLEV 16
#define TSIZE (1u << 19)
#define TMASK (TSIZE - 1u)
#define C0F 0.28209479177387814f

struct ResArgs { float res[NLEV]; };

// ---- WMMA helpers: f32 16x16x4, A = M x K slice, B = K x N slice ----
// A layout (32-bit, 16x4): lanes 0-15 -> M=lane, {K=0,K=1} in v0,v1 ; lanes 16-31 -> {K=2,K=3}
__device__ __forceinline__ v2f lda(const float (*act)[33], int tile, int ks, int ln) {
  int m  = ln & 15;
  int k2 = (ln >> 4) << 1;
  v2f a;
  a.x = act[tile * 16 + m][ks * 4 + k2];
  a.y = act[tile * 16 + m][ks * 4 + k2 + 1];
  return a;
}
// B layout (32-bit, 4x16): lanes 0-15 -> N=lane, {K=0,K=1}; lanes 16-31 -> {K=2,K=3}
__device__ __forceinline__ v2f ldb(const float (*w)[33], int nt, int ks, int ln) {
  int nn = ln & 15;
  int k2 = (ln >> 4) << 1;
  v2f b;
  b.x = w[ks * 4 + k2][nt * 16 + nn];
  b.y = w[ks * 4 + k2 + 1][nt * 16 + nn];
  return b;
}

// D(32x32) = A(32x32) @ B(32x32), optionally relu, via 2x2 tiles of 16x16, K in steps of 4.
__device__ __forceinline__ void gemm32x32(const float (*A)[33], const float (*B)[33],
                                          float (*D)[33], int ln, bool do_relu) {
#pragma unroll
  for (int tile = 0; tile < 2; ++tile) {
#pragma unroll
    for (int nt = 0; nt < 2; ++nt) {
      v8f acc = {0.f, 0.f, 0.f, 0.f, 0.f, 0.f, 0.f, 0.f};
#pragma unroll
      for (int ks = 0; ks < 8; ++ks) {
        v2f a = lda(A, tile, ks, ln);
        v2f b = ldb(B, nt, ks, ln);
        acc = __builtin_amdgcn_wmma_f32_16x16x4_f32(false, a, false, b, (short)0, acc,
                                                    false, false);
      }
      // D layout: VGPR r -> M = r (lanes 0-15) / r+8 (lanes 16-31), N = lane&15
      int col = ln & 15;
      int rb  = (ln >> 4) << 3;
#pragma unroll
      for (int r = 0; r < 8; ++r) {
        float v = acc[r];
        if (do_relu) v = fmaxf(v, 0.f);
        D[tile * 16 + rb + r][nt * 16 + col] = v;
      }
    }
  }
}

__global__ __launch_bounds__(128) void ngp_fused(
    const float* __restrict__ pos, const float* __restrict__ rx,
    const float* __restrict__ table, const float* __restrict__ W1,
    const float* __restrict__ W2, const float* __restrict__ aabb,
    const int* __restrict__ adeg, float* __restrict__ out, ResArgs ra, int npts) {
  __shared__ float w1s[32][33];
  __shared__ float w2s[32][33];
  __shared__ float w1ts[32][33];
  __shared__ float bufA[4][32][33];  // feats -> dH
  __shared__ float bufB[4][32][33];  // h -> dFeats
  __shared__ float bufC[4][32][33];  // sh

  const int tid = threadIdx.x;
  const int wv  = tid >> 5;
  const int ln  = tid & 31;
  const int n   = blockIdx.x * 128 + tid;
  const bool valid = (n < npts);
  const int nc  = valid ? n : (npts - 1);

  // Load weights (+ W1^T) into LDS
  for (int t = tid; t < 1024; t += 128) {
    int i = t >> 5, j = t & 31;
    float v1  = W1[t];
    w1s[i][j] = v1;
    w1ts[j][i] = v1;
    w2s[i][j] = W2[t];
  }
  __syncthreads();

  const float px = pos[nc * 3 + 0], py = pos[nc * 3 + 1], pz = pos[nc * 3 + 2];
  const float amn0 = aabb[0], amn1 = aabb[1], amn2 = aabb[2];
  const float sc0 = fmaxf(aabb[3] - amn0, 1e-6f);
  const float sc1 = fmaxf(aabb[4] - amn1, 1e-6f);
  const float sc2 = fmaxf(aabb[5] - amn2, 1e-6f);
  const float u0 = (px - amn0) / sc0, u1 = (py - amn1) / sc1, u2 = (pz - amn2) / sc2;
  const bool sel = (u0 >= 0.f) && (u0 <= 1.f) && (u1 >= 0.f) && (u1 <= 1.f) &&
                   (u2 >= 0.f) && (u2 <= 1.f);
  const float pn0 = fminf(fmaxf(u0, 0.f), 1.f);
  const float pn1 = fminf(fmaxf(u1, 0.f), 1.f);
  const float pn2 = fminf(fmaxf(u2, 0.f), 1.f);

  // ---- Phase A: hash encode -> feats in bufA[wv][ln][0..31] ----
  for (int l = 0; l < NLEV; ++l) {
    const float res = ra.res[l];
    const float fp0 = pn0 * res, fp1 = pn1 * res, fp2 = pn2 * res;
    const float fl0 = floorf(fp0), fl1 = floorf(fp1), fl2 = floorf(fp2);
    const float w0 = fp0 - fl0, w1w = fp1 - fl1, w2w = fp2 - fl2;
    const unsigned cx0 = (unsigned)fl0, cy0 = (unsigned)fl1, cz0 = (unsigned)fl2;
    const float* tl = table + (size_t)l * (size_t)TSIZE * 2u;
    float f0a = 0.f, f1a = 0.f;
#pragma unroll
    for (int i = 0; i < 8; ++i) {
      unsigned cx = cx0 + ((i >> 2) & 1), cy = cy0 + ((i >> 1) & 1), cz = cz0 + (i & 1);
      unsigned h   = cx ^ (cy * 2654435761u) ^ (cz * 805459861u);
      unsigned idx = h & TMASK;
      float2 f = *(const float2*)(tl + (size_t)idx * 2u);
      float t0 = (i & 4) ? w0 : 1.f - w0;
      float t1 = (i & 2) ? w1w : 1.f - w1w;
      float t2 = (i & 1) ? w2w : 1.f - w2w;
      float wc = t0 * t1 * t2;
      f0a += f.x * wc;
      f1a += f.y * wc;
    }
    bufA[wv][ln][2 * l]     = f0a;
    bufA[wv][ln][2 * l + 1] = f1a;
  }
  __syncthreads();

  // ---- Phase B/C: MLP forward via WMMA ----
  gemm32x32(bufA[wv], w1s, bufB[wv], ln, true);   // h = relu(feats @ W1)
  __syncthreads();
  gemm32x32(bufB[wv], w2s, bufC[wv], ln, false);  // sh = h @ W2
  __syncthreads();

  // ---- Phase D: per-point head + backprop seed dH -> bufA ----
  const float sh00 = bufC[wv][ln][0], sh01 = bufC[wv][ln][1];
  const float dens0 = C0F * sh00, dens1 = C0F * sh01;
  const float rr = sqrtf(dens0 * dens0 + dens1 * dens1);
  const float q0 = C0F * dens0 / rr;  // dL/d sh[0] ; rr==0 -> NaN (matches JAX, masked later)
  const float q1 = C0F * dens1 / rr;
  for (int j = 0; j < 32; ++j) {
    float dh = q0 * w2s[j][0] + q1 * w2s[j][1];
    float hj = bufB[wv][ln][j];
    bufA[wv][ln][j] = (hj > 0.f) ? dh : 0.f;  // relu'
  }
  __syncthreads();

  // ---- Phase E: dFeats = dH @ W1^T via WMMA -> bufB ----
  gemm32x32(bufA[wv], w1ts, bufB[wv], ln, false);
  __syncthreads();

  // ---- Phase F: gradient re-gather, normal, SH/scat, output ----
  float g0 = 0.f, g1 = 0.f, g2 = 0.f;
  for (int l = 0; l < NLEV; ++l) {
    const float res = ra.res[l];
    const float dfx = bufB[wv][ln][2 * l], dfy = bufB[wv][ln][2 * l + 1];
    const float fp0 = pn0 * res, fp1 = pn1 * res, fp2 = pn2 * res;
    const float fl0 = floorf(fp0), fl1 = floorf(fp1), fl2 = floorf(fp2);
    const float w0 = fp0 - fl0, w1w = fp1 - fl1, w2w = fp2 - fl2;
    const unsigned cx0 = (unsigned)fl0, cy0 = (unsigned)fl1, cz0 = (unsigned)fl2;
    const float* tl = table + (size_t)l * (size_t)TSIZE * 2u;
#pragma unroll
    for (int i = 0; i < 8; ++i) {
      unsigned cx = cx0 + ((i >> 2) & 1), cy = cy0 + ((i >> 1) & 1), cz = cz0 + (i & 1);
      unsigned h   = cx ^ (cy * 2654435761u) ^ (cz * 805459861u);
      unsigned idx = h & TMASK;
      float2 f = *(const float2*)(tl + (size_t)idx * 2u);
      float t0 = (i & 4) ? w0 : 1.f - w0;
      float t1 = (i & 2) ? w1w : 1.f - w1w;
      float t2 = (i & 1) ? w2w : 1.f - w2w;
      float coef = (dfx * f.x + dfy * f.y) * res;
      g0 += ((i & 4) ? coef : -coef) * t1 * t2;
      g1 += ((i & 2) ? coef : -coef) * t0 * t2;
      g2 += ((i & 1) ? coef : -coef) * t0 * t1;
    }
  }
  const float gw0 = g0 / sc0, gw1 = g1 / sc1, gw2 = g2 / sc2;
  const float gn = sqrtf(gw0 * gw0 + gw1 * gw1 + gw2 * gw2);
  const float dnm = fmaxf(gn, 1e-20f);
  float nr0 = -gw0 / dnm, nr1 = -gw1 / dnm, nr2 = -gw2 / dnm;
  nr0 = (nr0 != nr0) ? 0.f : nr0;
  nr1 = (nr1 != nr1) ? 0.f : nr1;
  nr2 = (nr2 != nr2) ? 0.f : nr2;

  // view dir + SH basis
  float ddx = rx[0] - px, ddy = rx[1] - py, ddz = rx[2] - pz;
  const float dl = fmaxf(sqrtf(ddx * ddx + ddy * ddy + ddz * ddz), 1e-20f);
  const float dx = ddx / dl, dy = ddy / dl, dz = ddz / dl;
  const float xx = dx * dx, yy = dy * dy, zz = dz * dz;
  const float xy = dx * dy, yz = dy * dz, xz = dx * dz;
  float Yv[16];
  Yv[0]  = 0.28209479177387814f;
  Yv[1]  = -0.4886025119029199f * dy;
  Yv[2]  = 0.4886025119029199f * dz;
  Yv[3]  = -0.4886025119029199f * dx;
  Yv[4]  = 1.0925484305920792f * xy;
  Yv[5]  = -1.0925484305920792f * yz;
  Yv[6]  = 0.31539156525252005f * (2.f * zz - xx - yy);
  Yv[7]  = -1.0925484305920792f * xz;
  Yv[8]  = 0.5462742152960396f * (xx - yy);
  Yv[9]  = -0.5900435899266435f * dy * (3.f * xx - yy);
  Yv[10] = 2.890611442640554f * xy * dz;
  Yv[11] = -0.4570457994644658f * dy * (4.f * zz - xx - yy);
  Yv[12] = 0.3731763325901154f * dz * (2.f * zz - 3.f * xx - 3.f * yy);
  Yv[13] = -0.4570457994644658f * dx * (4.f * zz - xx - yy);
  Yv[14] = 1.445305721320277f * dz * (xx - yy);
  Yv[15] = -0.5900435899266435f * dx * (xx - 3.f * yy);

  int mact = *adeg + 1;
  mact *= mact;
  float scat0 = 0.f, scat1 = 0.f;
  float* o = out + (size_t)n * 39u;
#pragma unroll
  for (int c = 0; c < 16; ++c) {
    float s0v = bufC[wv][ln][2 * c];
    float s1v = bufC[wv][ln][2 * c + 1];
    if (c < mact) {
      scat0 += Yv[c] * s0v;
      scat1 += Yv[c] * s1v;
    }
    if (valid) {
      o[7 + 2 * c] = sel ? s0v : 0.f;
      o[8 + 2 * c] = sel ? s1v : 0.f;
    }
  }
  if (valid) {
    o[0] = sel ? dens0 : 0.f;
    o[1] = sel ? dens1 : 0.f;
    o[2] = sel ? scat0 : 0.f;
    o[3] = sel ? scat1 : 0.f;
    o[4] = sel ? nr0 : 0.f;
    o[5] = sel ? nr1 : 0.f;
    o[6] = sel ? nr2 : 0.f;
  }
}

extern "C" void kernel_launch(void* const* d_in, const int* in_sizes, int n_in,
                              void* d_out, int out_size, void* d_ws, size_t ws_size,
                              hipStream_t stream) {
  const float* positions = (const float*)d_in[0];
  const float* rx        = (const float*)d_in[1];
  const float* table     = (const float*)d_in[2];
  const float* W1        = (const float*)d_in[3];
  const float* W2        = (const float*)d_in[4];
  const float* aabb      = (const float*)d_in[5];
  const int*   adeg      = (const int*)d_in[6];
  float* out = (float*)d_out;

  const int npts = in_sizes[0] / 3;

  ResArgs ra;
  const double s = exp((log(4096.0) - log(16.0)) / 15.0);
  for (int l = 0; l < NLEV; ++l) ra.res[l] = (float)(16.0 * pow(s, (double)l));

  const int blocks = (npts + 127) / 128;
  ngp_fused<<<blocks, 128, 0, stream>>>(positions, rx, table, W1, W2, aabb, adeg, out,
                                        ra, npts);
}